// GATLayer_39127152066975
// MI455X (gfx1250) — compile-verified
//
#include <hip/hip_runtime.h>
#include <hip/hip_bf16.h>
#include <math.h>

#define IN_DIM   128
#define OUT_DIM  48
#define NEG_SLOPE 0.01f

typedef __attribute__((ext_vector_type(2))) float v2f;
typedef __attribute__((ext_vector_type(8))) float v8f;

// ---------------------------------------------------------------------------
// Kernel 1: z = h @ W1^T via V_WMMA_F32_16X16X4_F32 (full fp32 precision).
// One wave per 16-row tile; 3 column tiles (48 = 3*16); W1 staged in LDS.
// A layout (16x4 f32): lanes 0-15 hold K=0,1 ; lanes 16-31 hold K=2,3.
// B layout (4x16 f32): mirrored; C/D: 8 VGPRs, M split across lane halves.
// ---------------------------------------------------------------------------
__global__ __launch_bounds__(256) void k_gemm(const float* __restrict__ h,
                                              const float* __restrict__ W1,
                                              float* __restrict__ z, int N) {
    __shared__ float sW1[OUT_DIM * IN_DIM];  // 24 KB
    for (int i = threadIdx.x; i < OUT_DIM * IN_DIM; i += 256)
        sW1[i] = W1[i];
    __syncthreads();

    const int wid  = threadIdx.x >> 5;   // wave id in block (0..7)
    const int lane = threadIdx.x & 31;
    const int m    = lane & 15;          // row-within-tile for A, col for B
    const int kh   = lane >> 4;          // K half (0: K=0,1 ; 1: K=2,3)
    const long tile = (long)blockIdx.x * 8 + wid;
    const long row0 = tile * 16;
    if (row0 >= N) return;               // wave-uniform: EXEC stays all-1s

    long arow = row0 + m;
    if (arow >= N) arow = N - 1;         // clamp loads; stores guarded below
    const float* hp  = h + arow * IN_DIM + 2 * kh;
    const float* b0p = &sW1[( 0 + m) * IN_DIM + 2 * kh];
    const float* b1p = &sW1[(16 + m) * IN_DIM + 2 * kh];
    const float* b2p = &sW1[(32 + m) * IN_DIM + 2 * kh];

    v8f acc0 = {}; v8f acc1 = {}; v8f acc2 = {};
    #pragma unroll 4
    for (int k = 0; k < IN_DIM; k += 4) {
        const v2f a  = *(const v2f*)(hp  + k);   // global_load_b64
        const v2f b0 = *(const v2f*)(b0p + k);   // ds_load_2addr_b64 (paired)
        const v2f b1 = *(const v2f*)(b1p + k);
        const v2f b2 = *(const v2f*)(b2p + k);
        acc0 = __builtin_amdgcn_wmma_f32_16x16x4_f32(false, a, false, b0, (short)0, acc0, false, false);
        acc1 = __builtin_amdgcn_wmma_f32_16x16x4_f32(false, a, false, b1, (short)0, acc1, false, false);
        acc2 = __builtin_amdgcn_wmma_f32_16x16x4_f32(false, a, false, b2, (short)0, acc2, false, false);
    }

    // C/D layout: VGPR j, lanes 0-15 -> M=j, lanes 16-31 -> M=j+8; N = m.
    if (row0 + 16 <= N) {
        // Full tile (common case): wave-uniform branch, straight-line stores.
        float* zp = z + (row0 + 8 * kh) * OUT_DIM + m;
        #pragma unroll
        for (int j = 0; j < 8; ++j) {
            zp[j * OUT_DIM +  0] = acc0[j];
            zp[j * OUT_DIM + 16] = acc1[j];
            zp[j * OUT_DIM + 32] = acc2[j];
        }
    } else {
        // Partial tail tile: per-row guard.
        #pragma unroll
        for (int j = 0; j < 8; ++j) {
            long row = row0 + j + 8 * kh;
            if (row < N) {
                float* zp = z + row * OUT_DIM + m;
                zp[0]  = acc0[j];
                zp[16] = acc1[j];
                zp[32] = acc2[j];
            }
        }
    }
}

// ---------------------------------------------------------------------------
// Kernel 2: per-node attention scores es = z . Wa, ed = z . Wb
// ---------------------------------------------------------------------------
__global__ void k_scores(const float* __restrict__ z, const float* __restrict__ W2,
                         float* __restrict__ es, float* __restrict__ ed, int N) {
    int i = blockIdx.x * blockDim.x + threadIdx.x;
    if (i >= N) return;
    const float* zr = z + (long)i * OUT_DIM;
    float a = 0.f, b = 0.f;
    #pragma unroll
    for (int n = 0; n < OUT_DIM; ++n) {
        float v = zr[n];
        a += v * W2[n];            // Wa (uniform address -> scalar cached)
        b += v * W2[OUT_DIM + n];  // Wb
    }
    es[i] = a;
    ed[i] = b;
}

// ---------------------------------------------------------------------------
// Kernel 3: init (runs every call -> graph-replay deterministic)
// ---------------------------------------------------------------------------
__global__ void k_init(float* __restrict__ out, float* __restrict__ emax,
                       float* __restrict__ denom, int N) {
    long idx = (long)blockIdx.x * blockDim.x + threadIdx.x;
    long total = (long)N * OUT_DIM;
    if (idx < total) out[idx] = 0.f;
    if (idx < N) {
        emax[idx]  = __int_as_float(0xFF800000); // -inf
        denom[idx] = 0.f;
    }
}

// float atomic max via sign-split (works with -inf init)
__device__ inline void atomicMaxFloat(float* addr, float val) {
    if (val >= 0.f)
        atomicMax((int*)addr, __float_as_int(val));
    else
        atomicMin((unsigned int*)addr, __float_as_uint(val));
}

// ---------------------------------------------------------------------------
// Kernel 4: edge logits + segment max
// ---------------------------------------------------------------------------
__global__ void k_edge_max(const float* __restrict__ es, const float* __restrict__ ed,
                           const int* __restrict__ src, const int* __restrict__ dst,
                           float* __restrict__ e, float* __restrict__ emax, int E) {
    int i = blockIdx.x * blockDim.x + threadIdx.x;
    if (i >= E) return;
    float v = es[src[i]] + ed[dst[i]];
    v = v > 0.f ? v : NEG_SLOPE * v;     // leaky relu
    e[i] = v;
    atomicMaxFloat(&emax[dst[i]], v);
}

// ---------------------------------------------------------------------------
// Kernel 5: ex = exp(e - emax[dst]); denom += ex (in-place rewrite of e)
// ---------------------------------------------------------------------------
__global__ void k_edge_exp(const int* __restrict__ dst, float* __restrict__ e,
                           const float* __restrict__ emax, float* __restrict__ denom,
                           int E) {
    int i = blockIdx.x * blockDim.x + threadIdx.x;
    if (i >= E) return;
    int d = dst[i];
    float ex = expf(e[i] - emax[d]);
    e[i] = ex;
    atomicAdd(&denom[d], ex);
}

// ---------------------------------------------------------------------------
// Kernel 6: out[dst] += (ex/denom[dst]) * z[src]
// 64-thread groups per edge (48 active lanes) -> coalesced row gather/atomics
// ---------------------------------------------------------------------------
__global__ void k_agg(const int* __restrict__ src, const int* __restrict__ dst,
                      const float* __restrict__ e, const float* __restrict__ denom,
                      const float* __restrict__ z, float* __restrict__ out, int E) {
    int tl  = threadIdx.x & 63;  // lane within edge group
    int sub = threadIdx.x >> 6;  // edge group within block (0..3)
    long edge = (long)blockIdx.x * 4 + sub;
    if (edge >= E || tl >= OUT_DIM) return;
    int s = src[edge], d = dst[edge];
    float alpha = e[edge] / denom[d];
    atomicAdd(&out[(long)d * OUT_DIM + tl], alpha * z[(long)s * OUT_DIM + tl]);
}

// ---------------------------------------------------------------------------
// Kernel 7: nodes with zero in-degree (emax still -inf) copy z through
// ---------------------------------------------------------------------------
__global__ void k_fix(const float* __restrict__ z, const float* __restrict__ emax,
                      float* __restrict__ out, int N) {
    long idx = (long)blockIdx.x * blockDim.x + threadIdx.x;
    if (idx >= (long)N * OUT_DIM) return;
    int node = (int)(idx / OUT_DIM);
    if (__float_as_uint(emax[node]) == 0xFF800000u)  // still -inf => deg 0
        out[idx] = z[idx];
}

// ---------------------------------------------------------------------------
extern "C" void kernel_launch(void* const* d_in, const int* in_sizes, int n_in,
                              void* d_out, int out_size, void* d_ws, size_t ws_size,
                              hipStream_t stream) {
    const float* h   = (const float*)d_in[0];
    const float* W1  = (const float*)d_in[1];
    const float* W2  = (const float*)d_in[2];
    const int*   src = (const int*)d_in[3];
    const int*   dst = (const int*)d_in[4];
    const int N = in_sizes[0] / IN_DIM;
    const int E = in_sizes[3];

    float* ws    = (float*)d_ws;
    float* z     = ws;                       // N*48
    float* es    = z + (size_t)N * OUT_DIM;  // N
    float* ed    = es + N;                   // N
    float* emax  = ed + N;                   // N
    float* denom = emax + N;                 // N
    float* e     = denom + N;                // E
    float* out   = (float*)d_out;

    const int tiles = (N + 15) / 16;
    k_gemm  <<<(tiles + 7) / 8,                     256, 0, stream>>>(h, W1, z, N);
    k_scores<<<(N + 255) / 256,                     256, 0, stream>>>(z, W2, es, ed, N);
    k_init  <<<((long)N * OUT_DIM + 255) / 256,     256, 0, stream>>>(out, emax, denom, N);
    k_edge_max<<<(E + 255) / 256,                   256, 0, stream>>>(es, ed, src, dst, e, emax, E);
    k_edge_exp<<<(E + 255) / 256,                   256, 0, stream>>>(dst, e, emax, denom, E);
    k_agg   <<<((long)E + 3) / 4,                   256, 0, stream>>>(src, dst, e, denom, z, out, E);
    k_fix   <<<((long)N * OUT_DIM + 255) / 256,     256, 0, stream>>>(z, emax, out, N);
}